// cheb_conv_16260746183116
// MI455X (gfx1250) — compile-verified
//
#include <hip/hip_runtime.h>
#include <hip/hip_bf16.h>
#include <stdint.h>

// Problem constants (match reference)
#define NB    16          // batch
#define NN    50000       // nodes
#define NFIN  32
#define NFOUT 32
#define NK    6
#define NNZ_  800000
#define ND    512         // NB * NFIN, per-node state width
#define NROWS (NB * NN)   // 800000 output rows

typedef __attribute__((ext_vector_type(2))) float v2f;
typedef __attribute__((ext_vector_type(8))) float v8f;

// ---------------------------------------------------------------------------
// CSR construction: histogram -> exclusive scan -> scatter
// ---------------------------------------------------------------------------
__global__ void k_hist(const int* __restrict__ rows, int* __restrict__ cnt, int nnz) {
    int i = blockIdx.x * blockDim.x + threadIdx.x;
    if (i < nnz) atomicAdd(&cnt[rows[i]], 1);
}

// single-block chunked exclusive scan over n counts -> row_ptr[0..n]
__global__ void k_scan(const int* __restrict__ cnt, int* __restrict__ row_ptr, int n) {
    __shared__ int sh[1024];
    __shared__ int carry;
    int tid = threadIdx.x;
    if (tid == 0) carry = 0;
    __syncthreads();
    for (int base = 0; base < n; base += 1024) {
        int i = base + tid;
        int v = (i < n) ? cnt[i] : 0;
        sh[tid] = v;
        __syncthreads();
        for (int off = 1; off < 1024; off <<= 1) {
            int t = (tid >= off) ? sh[tid - off] : 0;
            __syncthreads();
            sh[tid] += t;
            __syncthreads();
        }
        int incl = sh[tid];
        int excl = incl - v;
        int c = carry;
        if (i < n) row_ptr[i] = c + excl;
        __syncthreads();
        if (tid == 1023) carry = c + incl;
        __syncthreads();
    }
    if (tid == 0) row_ptr[n] = carry;
}

__global__ void k_scatter(const int* __restrict__ rows, const int* __restrict__ cols,
                          const float* __restrict__ vals,
                          const int* __restrict__ row_ptr, int* __restrict__ fill,
                          int* __restrict__ ecol, float* __restrict__ eval, int nnz) {
    int i = blockIdx.x * blockDim.x + threadIdx.x;
    if (i < nnz) {
        int r = rows[i];
        int p = row_ptr[r] + atomicAdd(&fill[r], 1);
        ecol[p] = cols[i];
        eval[p] = vals[i];
    }
}

// ---------------------------------------------------------------------------
// T0[n, b*32+f] = x[b, n, f]   (state layout: [N, 512], b-major cols)
// ---------------------------------------------------------------------------
__global__ void k_transpose(const float* __restrict__ x, float* __restrict__ T0) {
    int idx = blockIdx.x * blockDim.x + threadIdx.x;   // over N*128 float4 slots
    int n = idx >> 7;          // 128 float4 per node row
    int q = idx & 127;         // column-quad: b = q/8, f4 = q%8
    int b = q >> 3, fq = q & 7;
    const float4 v = *(const float4*)(x + ((size_t)b * NN + n) * NFIN + fq * 4);
    *(float4*)(T0 + (size_t)n * ND + q * 4) = v;
}

// ---------------------------------------------------------------------------
// Gather SpMM: Tout[n,:] = alpha * sum_e val[e]*Tin[col[e],:] + beta * Tout_old[n,:]
// one 128-thread block per row, float4 per lane (2KB coalesced row IO).
// Edge (col,val) chunks are staged in LDS so the FMA loop does exactly one
// global load (the 16B gather) per edge per lane; edge metadata comes from
// LDS broadcast reads.
// ---------------------------------------------------------------------------
__global__ void k_spmm(float* __restrict__ Tout, const float* __restrict__ Tin,
                       const int* __restrict__ row_ptr,
                       const int* __restrict__ ecol, const float* __restrict__ eval,
                       float alpha, float beta) {
    __shared__ int   s_col[128];
    __shared__ float s_val[128];
    const int n = blockIdx.x;
    const int t = threadIdx.x;
    const int d = t * 4;
    const int e0 = row_ptr[n], e1 = row_ptr[n + 1];
    float4 acc = make_float4(0.f, 0.f, 0.f, 0.f);
    for (int base = e0; base < e1; base += 128) {
        const int m = (e1 - base < 128) ? (e1 - base) : 128;
        __syncthreads();                        // protect LDS reuse across chunks
        if (t < m) {
            s_col[t] = ecol[base + t];
            s_val[t] = eval[base + t];
        }
        __syncthreads();
        for (int j = 0; j < m; ++j) {
            const int   c = s_col[j];           // LDS broadcast
            const float v = s_val[j];
            const float4 tt = *(const float4*)(Tin + (size_t)c * ND + d);
            acc.x = fmaf(v, tt.x, acc.x);
            acc.y = fmaf(v, tt.y, acc.y);
            acc.z = fmaf(v, tt.z, acc.z);
            acc.w = fmaf(v, tt.w, acc.w);
        }
    }
    float* op = Tout + (size_t)n * ND + d;
    float4 r;
    if (beta != 0.f) {
        const float4 prev = *(const float4*)op;
        r.x = fmaf(alpha, acc.x, beta * prev.x);
        r.y = fmaf(alpha, acc.y, beta * prev.y);
        r.z = fmaf(alpha, acc.z, beta * prev.z);
        r.w = fmaf(alpha, acc.w, beta * prev.w);
    } else {
        r.x = alpha * acc.x; r.y = alpha * acc.y;
        r.z = alpha * acc.z; r.w = alpha * acc.w;
    }
    *(float4*)op = r;
}

// ---------------------------------------------------------------------------
// Dense accumulation with WMMA fp32:
//   out[r, fo] (+)= sum_f Tk[n, b*32+f] * W[(f*6+k), fo],  r = b*N + n
// 256 threads = 8 waves = 4 row-tiles x 2 col-tiles; 64 rows per block.
// N % 16 == 0, so each 16-row tile lies within one batch b.
// All `out` indices fit in int (<= 25.6M elements) -> immediate-offset stores.
// ---------------------------------------------------------------------------
__global__ void k_gemm_acc(const float* __restrict__ Tk, const float* __restrict__ W,
                           float* __restrict__ out, int k, int accflag) {
    __shared__ float shT[4][16][33];   // padded: row stride 33 dwords
    __shared__ float shW[32][33];
    const int tid = threadIdx.x;
    const int r0 = blockIdx.x * 64;

    // stage W_k (32x32) into LDS
    for (int i = tid; i < 1024; i += 256) {
        int f = i >> 5, fo = i & 31;
        shW[f][fo] = W[(f * NK + k) * NFOUT + fo];
    }
    // stage 64 node-rows x 32 features into LDS (coalesced 128B per row)
    for (int i = tid; i < 2048; i += 256) {
        int row = i >> 5, col = i & 31;
        int r = r0 + row;
        int b = r / NN, n = r % NN;
        shT[row >> 4][row & 15][col] = Tk[(size_t)n * ND + b * NFIN + col];
    }
    __syncthreads();

    const int wave = tid >> 5, lane = tid & 31;
    const int tile = wave >> 1;              // 0..3 row tile
    const int colt = (wave & 1) * 16;        // 0 or 16
    const int lm = lane & 15, lh = lane >> 4;

    // preload all 8 B fragments (4x16 K-slabs of W_k) to registers
    v2f bfrag[8];
#pragma unroll
    for (int kk = 0; kk < 8; ++kk) {
        int K2 = kk * 4 + lh * 2;
        bfrag[kk].x = shW[K2][colt + lm];
        bfrag[kk].y = shW[K2 + 1][colt + lm];
    }

    const int obase = (r0 + tile * 16 + lh * 8) * NFOUT + colt + lm;
    v8f c = {};
    if (accflag) {
#pragma unroll
        for (int r = 0; r < 8; ++r)
            c[r] = out[obase + r * NFOUT];
    }

#pragma unroll
    for (int kk = 0; kk < 8; ++kk) {
        v2f a;
        int K2 = kk * 4 + lh * 2;
        a.x = shT[tile][lm][K2];
        a.y = shT[tile][lm][K2 + 1];
        c = __builtin_amdgcn_wmma_f32_16x16x4_f32(
                /*neg_a=*/false, a, /*neg_b=*/false, bfrag[kk],
                /*c_mod=*/(short)0, c, /*reuse_a=*/false, /*reuse_b=*/false);
    }

#pragma unroll
    for (int r = 0; r < 8; ++r)
        out[obase + r * NFOUT] = c[r];
}

// ---------------------------------------------------------------------------
// Launch
// ---------------------------------------------------------------------------
static inline size_t align256(size_t v) { return (v + 255) & ~(size_t)255; }

extern "C" void kernel_launch(void* const* d_in, const int* in_sizes, int n_in,
                              void* d_out, int out_size, void* d_ws, size_t ws_size,
                              hipStream_t stream) {
    (void)in_sizes; (void)n_in; (void)out_size; (void)ws_size;
    const float* x    = (const float*)d_in[0];
    const int*   rows = (const int*)d_in[1];
    const int*   cols = (const int*)d_in[2];
    const float* vals = (const float*)d_in[3];
    const float* W    = (const float*)d_in[4];
    float* out = (float*)d_out;

    // workspace carve (~211.6 MB)
    char* ws = (char*)d_ws;
    size_t off = 0;
    const size_t SZT = (size_t)NN * ND * sizeof(float);        // 102.4 MB
    float* Ta = (float*)(ws + off); off = align256(off + SZT);
    float* Tb = (float*)(ws + off); off = align256(off + SZT);
    int* row_ptr = (int*)(ws + off); off = align256(off + (NN + 1) * sizeof(int));
    int* row_tmp = (int*)(ws + off); off = align256(off + NN * sizeof(int));
    int* ecol = (int*)(ws + off);    off = align256(off + (size_t)NNZ_ * sizeof(int));
    float* eval = (float*)(ws + off); off = align256(off + (size_t)NNZ_ * sizeof(float));

    // --- CSR build ---
    hipMemsetAsync(row_tmp, 0, NN * sizeof(int), stream);
    k_hist<<<(NNZ_ + 255) / 256, 256, 0, stream>>>(rows, row_tmp, NNZ_);
    k_scan<<<1, 1024, 0, stream>>>(row_tmp, row_ptr, NN);
    hipMemsetAsync(row_tmp, 0, NN * sizeof(int), stream);
    k_scatter<<<(NNZ_ + 255) / 256, 256, 0, stream>>>(rows, cols, vals, row_ptr,
                                                      row_tmp, ecol, eval, NNZ_);

    // --- T0 = transposed x ---
    k_transpose<<<(NN * 128) / 256, 256, 0, stream>>>(x, Ta);

    const int gGemm = NROWS / 64;   // 12500
    // k = 0 : out = T0 @ W_0   (initializes out)
    k_gemm_acc<<<gGemm, 256, 0, stream>>>(Ta, W, out, 0, 0);

    // k = 1 : T1 = L @ T0
    k_spmm<<<NN, 128, 0, stream>>>(Tb, Ta, row_ptr, ecol, eval, 1.f, 0.f);
    k_gemm_acc<<<gGemm, 256, 0, stream>>>(Tb, W, out, 1, 1);

    // k = 2..5 : T_k = 2 L T_{k-1} - T_{k-2}, written in place over T_{k-2}
    float* prev = Ta;   // holds T_{k-2}
    float* cur  = Tb;   // holds T_{k-1}
    for (int k = 2; k < NK; ++k) {
        k_spmm<<<NN, 128, 0, stream>>>(prev, cur, row_ptr, ecol, eval, 2.f, -1.f);
        k_gemm_acc<<<gGemm, 256, 0, stream>>>(prev, W, out, k, 1);
        float* t = prev; prev = cur; cur = t;
    }
}